// SegmentationLoss_35854386987723
// MI455X (gfx1250) — compile-verified
//
#include <hip/hip_runtime.h>
#include <math.h>

#define NUM_CLASSES 172
#define IGNORE_INDEX 255
#define BATCH 4
#define HW 65536               // 256*256
#define NPIX (BATCH * HW)      // 262144 pixels total
#define DICE_W 0.5f
#define EPS_F 1e-7f

typedef float v2f __attribute__((ext_vector_type(2)));
typedef float v8f __attribute__((ext_vector_type(8)));

// ---------------- workspace layout (floats) ----------------
// [0, NPIX)                 : logZ per pixel (max + log sum exp)
// [NPIX + 0*C, +1*C)        : sumP[c]   (sum of softmax prob over all pixels)
// [NPIX + 1*C, +2*C)        : inter[c]  (sum of prob where target==c)
// [NPIX + 2*C, +3*C)        : cnt[c]    (count of pixels with target==c)
// [NPIX + 3*C]              : ceSum
// [NPIX + 3*C + 1]          : nValid

__global__ __launch_bounds__(256) void init_kernel(float* __restrict__ acc) {
    int i = threadIdx.x + blockIdx.x * blockDim.x;
    if (i < 3 * NUM_CLASSES + 2) acc[i] = 0.0f;
}

// Kernel 1: per-pixel logZ + CE partials. One wave owns 64 contiguous pixels.
// Phase 1: float2 (b64) coalesced max/gather pass (256B per wave per class).
// Phase 2: sum_c exp(x-m) via V_WMMA_F32_16X16X4_F32, A = ones(16x4),
//          B = exp-values (K=4 classes x N=16 pixels), 4 accumulator groups.
__global__ __launch_bounds__(256) void logz_ce_kernel(const float* __restrict__ pred,
                                                      const int*   __restrict__ target,
                                                      float* __restrict__ logZ,
                                                      float* __restrict__ ceSum,
                                                      float* __restrict__ nValid) {
    const int lane = threadIdx.x & 31;
    const int gw   = (blockIdx.x * blockDim.x + threadIdx.x) >> 5;  // global wave id
    const int pix0 = gw << 6;                                       // 64 pixels per wave
    const int bImg = pix0 >> 16;                                    // pix0 / HW
    const int hw0  = pix0 & (HW - 1);

    // ---- phase 1: per-lane 2 pixels (2L, 2L+1): max over classes + x[target] ----
    const int2 tg = ((const int2*)target)[(gw << 5) + lane];
    float m0 = -3.4e38f, m1 = -3.4e38f;
    float xt0 = 0.0f, xt1 = 0.0f;
    const float2* pb = (const float2*)(pred + (size_t)bImg * NUM_CLASSES * HW + hw0) + lane;
    #pragma unroll 4
    for (int c = 0; c < NUM_CLASSES; ++c) {
        const float2 x = pb[(size_t)c * (HW / 2)];
        m0 = fmaxf(m0, x.x);
        m1 = fmaxf(m1, x.y);
        if (c == tg.x) xt0 = x.x;
        if (c == tg.y) xt1 = x.y;
    }

    // ---- broadcast per-group maxima (group g = pixels g*16 .. g*16+15) ----
    // pixel q = g*16 + (lane&15) is owned by lane q>>1 = g*8 + ((lane&15)>>1), slot q&1 = lane&1
    const int nIdx = lane & 15;
    float mg[4];
    #pragma unroll
    for (int g = 0; g < 4; ++g) {
        const int src = (g << 3) + (nIdx >> 1);
        const float ma = __shfl(m0, src);
        const float mb = __shfl(m1, src);
        mg[g] = (lane & 1) ? mb : ma;
    }

    // ---- phase 2: WMMA reduction over classes (172 = 43*4, exact) ----
    const int cOff = (lane >> 4) << 1;   // K sub-offset: 0 (lanes<16) or 2 (lanes>=16)
    const float* pgBase = pred + (size_t)bImg * NUM_CLASSES * HW + hw0 + nIdx;
    v2f aOnes; aOnes[0] = 1.0f; aOnes[1] = 1.0f;
    v8f acc0 = {}, acc1 = {}, acc2 = {}, acc3 = {};
    for (int ck = 0; ck < NUM_CLASSES; ck += 4) {
        const int    cA = ck + cOff;
        const float* r0 = pgBase + (size_t)cA * HW;
        const float* r1 = pgBase + (size_t)(cA + 1) * HW;
        v2f b0, b1, b2, b3;
        b0[0] = __expf(r0[ 0] - mg[0]);  b0[1] = __expf(r1[ 0] - mg[0]);
        b1[0] = __expf(r0[16] - mg[1]);  b1[1] = __expf(r1[16] - mg[1]);
        b2[0] = __expf(r0[32] - mg[2]);  b2[1] = __expf(r1[32] - mg[2]);
        b3[0] = __expf(r0[48] - mg[3]);  b3[1] = __expf(r1[48] - mg[3]);
        acc0 = __builtin_amdgcn_wmma_f32_16x16x4_f32(false, aOnes, false, b0, (short)0, acc0, false, false);
        acc1 = __builtin_amdgcn_wmma_f32_16x16x4_f32(false, aOnes, false, b1, (short)0, acc1, false, false);
        acc2 = __builtin_amdgcn_wmma_f32_16x16x4_f32(false, aOnes, false, b2, (short)0, acc2, false, false);
        acc3 = __builtin_amdgcn_wmma_f32_16x16x4_f32(false, aOnes, false, b3, (short)0, acc3, false, false);
    }

    // ---- gather per-pixel sums back to owner lanes ----
    // lane L owns pixels q0=2L, q1=2L+1; both in group g = L>>3, columns n0=(2L)&15, n0+1.
    const int n0 = (lane << 1) & 15;
    const int gSel = lane >> 3;
    const float t0 = __shfl(acc0[0], n0),     t1 = __shfl(acc1[0], n0);
    const float t2 = __shfl(acc2[0], n0),     t3 = __shfl(acc3[0], n0);
    const float u0 = __shfl(acc0[0], n0 + 1), u1 = __shfl(acc1[0], n0 + 1);
    const float u2 = __shfl(acc2[0], n0 + 1), u3 = __shfl(acc3[0], n0 + 1);
    const float S0 = (gSel == 0) ? t0 : (gSel == 1) ? t1 : (gSel == 2) ? t2 : t3;
    const float S1 = (gSel == 0) ? u0 : (gSel == 1) ? u1 : (gSel == 2) ? u2 : u3;

    const float lz0 = m0 + __logf(S0);
    const float lz1 = m1 + __logf(S1);
    float2 lzv; lzv.x = lz0; lzv.y = lz1;
    ((float2*)logZ)[(gw << 5) + lane] = lzv;

    // ---- CE partials (mean NLL over valid pixels) ----
    const bool v0 = (tg.x != IGNORE_INDEX);
    const bool v1 = (tg.y != IGNORE_INDEX);
    float nll = (v0 ? (lz0 - xt0) : 0.0f) + (v1 ? (lz1 - xt1) : 0.0f);
    float cnt = (v0 ? 1.0f : 0.0f) + (v1 ? 1.0f : 0.0f);
    #pragma unroll
    for (int o = 16; o > 0; o >>= 1) {
        nll += __shfl_xor(nll, o);
        cnt += __shfl_xor(cnt, o);
    }
    if (lane == 0) {
        atomicAdd(ceSum, nll);
        atomicAdd(nValid, cnt);
    }
}

// Kernel 2: per-class sums. Block = (8192-pixel chunk) x (class); thread owns 4
// consecutive pixels (b128 loads). pred should be L2-resident (180MB vs 192MB L2).
__global__ __launch_bounds__(256) void class_sum_kernel(const float* __restrict__ pred,
                                                        const int*   __restrict__ target,
                                                        const float* __restrict__ logZ,
                                                        float* __restrict__ sumP,
                                                        float* __restrict__ inter,
                                                        float* __restrict__ cnt) {
    const int c    = blockIdx.y;
    const int p0   = blockIdx.x * 8192;        // chunk never crosses a batch image
    const int bImg = p0 >> 16;
    const int hw0  = p0 & (HW - 1);
    const int t    = threadIdx.x;
    const float4* pc4 = (const float4*)(pred + (size_t)(bImg * NUM_CLASSES + c) * HW + hw0);
    const float4* lz4 = (const float4*)(logZ + p0);
    const int4*   tg4 = (const int4*)(target + p0);

    float aU = 0.0f, aI = 0.0f, aN = 0.0f;
    #pragma unroll 2
    for (int it = 0; it < 8; ++it) {
        const int o = t + it * 256;
        __builtin_prefetch(&pc4[o + 512], 0, 1);            // global_prefetch_b8
        const float4 x  = pc4[o];
        const float4 lz = lz4[o];
        const int4   tv = tg4[o];
        const float px = __expf(x.x - lz.x);
        const float py = __expf(x.y - lz.y);
        const float pz = __expf(x.z - lz.z);
        const float pw = __expf(x.w - lz.w);
        aU += (px + py) + (pz + pw);
        if (tv.x == c) { aI += px; aN += 1.0f; }
        if (tv.y == c) { aI += py; aN += 1.0f; }
        if (tv.z == c) { aI += pz; aN += 1.0f; }
        if (tv.w == c) { aI += pw; aN += 1.0f; }
    }
    #pragma unroll
    for (int o = 16; o > 0; o >>= 1) {
        aU += __shfl_xor(aU, o);
        aI += __shfl_xor(aI, o);
        aN += __shfl_xor(aN, o);
    }
    __shared__ float sU, sI, sN;
    if (t == 0) { sU = 0.0f; sI = 0.0f; sN = 0.0f; }
    __syncthreads();
    if ((t & 31) == 0) { atomicAdd(&sU, aU); atomicAdd(&sI, aI); atomicAdd(&sN, aN); }
    __syncthreads();
    if (t == 0) {
        atomicAdd(&sumP[c], sU);
        atomicAdd(&inter[c], sI);
        atomicAdd(&cnt[c],  sN);
    }
}

// Kernel 3: combine 172 dice terms + CE into the scalar output.
__global__ __launch_bounds__(256) void finalize_kernel(const float* __restrict__ sumP,
                                                       const float* __restrict__ inter,
                                                       const float* __restrict__ cnt,
                                                       const float* __restrict__ ceSum,
                                                       const float* __restrict__ nValid,
                                                       float* __restrict__ out) {
    __shared__ float sTerms, sValid;
    const int t = threadIdx.x;
    if (t == 0) { sTerms = 0.0f; sValid = 0.0f; }
    __syncthreads();
    float term = 0.0f, vc = 0.0f;
    if (t < NUM_CLASSES) {
        const float u = sumP[t] + cnt[t];          // union = sum(p) + counts
        if (u > 0.0f) {
            term = (2.0f * inter[t] + EPS_F) / (u + EPS_F);
            vc = 1.0f;
        }
    }
    #pragma unroll
    for (int o = 16; o > 0; o >>= 1) {
        term += __shfl_xor(term, o);
        vc   += __shfl_xor(vc, o);
    }
    if ((t & 31) == 0) { atomicAdd(&sTerms, term); atomicAdd(&sValid, vc); }
    __syncthreads();
    if (t == 0) {
        const float nvc  = sValid;
        const float dice = (nvc > 0.0f) ? (1.0f - sTerms / fmaxf(nvc, 1.0f)) : 0.0f;
        const float ce   = ceSum[0] / fmaxf(nValid[0], 1.0f);
        out[0] = ce + DICE_W * dice;
    }
}

extern "C" void kernel_launch(void* const* d_in, const int* in_sizes, int n_in,
                              void* d_out, int out_size, void* d_ws, size_t ws_size,
                              hipStream_t stream) {
    const float* pred   = (const float*)d_in[0];
    const int*   target = (const int*)d_in[1];

    float* ws     = (float*)d_ws;
    float* logZ   = ws;
    float* acc    = ws + NPIX;
    float* sumP   = acc;
    float* inter  = acc + NUM_CLASSES;
    float* cnt    = acc + 2 * NUM_CLASSES;
    float* ceSum  = acc + 3 * NUM_CLASSES;
    float* nValid = acc + 3 * NUM_CLASSES + 1;

    init_kernel<<<3, 256, 0, stream>>>(acc);
    // 64 pixels per wave, 8 waves per block -> 512 pixels per block
    logz_ce_kernel<<<NPIX / 512, 256, 0, stream>>>(pred, target, logZ, ceSum, nValid);
    class_sum_kernel<<<dim3(NPIX / 8192, NUM_CLASSES), 256, 0, stream>>>(
        pred, target, logZ, sumP, inter, cnt);
    finalize_kernel<<<1, 256, 0, stream>>>(sumP, inter, cnt, ceSum, nValid, (float*)d_out);
}